// ConvolutionHyperbolic_70480413327940
// MI455X (gfx1250) — compile-verified
//
#include <hip/hip_runtime.h>
#include <math.h>

typedef float v2f __attribute__((ext_vector_type(2)));
typedef float v8f __attribute__((ext_vector_type(8)));

#define NROW 1024
#define DIM  128
#define FEPS 1e-15f
#define MAXNORM 0.999f           // (1 - PROJ_EPS)/sqrt(C), C = 1
#define ATCLIP (1.0f - 1e-5f)    // artanh clip

// ---------- wave32 helpers ----------
__device__ __forceinline__ float waveSum(float v) {
#pragma unroll
  for (int m = 16; m >= 1; m >>= 1) v += __shfl_xor(v, m, 32);
  return v;
}

__device__ __forceinline__ float artanh_c(float x) {
  x = fminf(fmaxf(x, -ATCLIP), ATCLIP);
  return 0.5f * logf((1.0f + x) / (1.0f - x));
}

__device__ __forceinline__ float dot4(float4 a, float4 b) {
  return a.x * b.x + a.y * b.y + a.z * b.z + a.w * b.w;
}

// ---------- stage 1: xp = proj(x), xn = clip(||xp||, EPS) ----------
__global__ __launch_bounds__(32)
void k_proj_x(const float* __restrict__ x, float* __restrict__ xp,
              float* __restrict__ xn) {
  const int row = blockIdx.x, lane = threadIdx.x;
  float4 v = ((const float4*)(x + row * DIM))[lane];
  float n = fmaxf(sqrtf(waveSum(dot4(v, v))), FEPS);
  float s = (n > MAXNORM) ? (MAXNORM / n) : 1.0f;
  v.x *= s; v.y *= s; v.z *= s; v.w *= s;
  ((float4*)(xp + row * DIM))[lane] = v;
  if (lane == 0) xn[row] = fmaxf(fminf(n, MAXNORM), FEPS);
}

// ---------- stage 2: eb = proj(expmap0(b)) ----------
__global__ __launch_bounds__(32)
void k_bias(const float* __restrict__ b, float* __restrict__ eb) {
  const int lane = threadIdx.x;
  float4 v = ((const float4*)b)[lane];
  float n = fmaxf(sqrtf(waveSum(dot4(v, v))), FEPS);
  float t = tanhf(n) / n;                       // expmap0 scale
  v.x *= t; v.y *= t; v.z *= t; v.w *= t;
  float ne = fmaxf(sqrtf(waveSum(dot4(v, v))), FEPS);
  float s = (ne > MAXNORM) ? (MAXNORM / ne) : 1.0f;
  v.x *= s; v.y *= s; v.z *= s; v.w *= s;
  ((float4*)eb)[lane] = v;
}

// ---------- fp32 WMMA GEMM, C[M,N] = A[M,K] * Bt[N,K]^T  (both row-major) ----
// One wave computes a 16 x (16*NT) strip. Per unrolled K-chunk all fragments
// are staged into registers FIRST (one load clause), then U*NT WMMAs issue,
// giving progressive s_wait_loadcnt instead of wait-0 per WMMA.
// fp32 16x4 A layout: lanes 0-15 <-> M, lanes 16-31 hold K-offset +2;
// v2f = {K, K+1}. B fragment mirrors it with lanes <-> N.
template <int NT, int U>
__global__ __launch_bounds__(32)
void k_gemm_nt(const float* __restrict__ A, const float* __restrict__ Bt,
               float* __restrict__ C, int K, int ldc) {
  const int j0 = blockIdx.x * (16 * NT);
  const int i0 = blockIdx.y * 16;
  const int lane = threadIdx.x;
  const int m = lane & 15;
  const int koff = (lane >> 4) * 2;
  const float* arow = A + (size_t)(i0 + m) * K + koff;
  const float* brow = Bt + (size_t)(j0 + m) * K + koff;
  v8f acc[NT];
#pragma unroll
  for (int t = 0; t < NT; ++t) acc[t] = (v8f){0.f,0.f,0.f,0.f,0.f,0.f,0.f,0.f};
  for (int k = 0; k < K; k += 4 * U) {
    v2f a[U];
    v2f bf[U][NT];
#pragma unroll
    for (int u = 0; u < U; ++u) {
      a[u] = *(const v2f*)(arow + k + 4 * u);
#pragma unroll
      for (int t = 0; t < NT; ++t)
        bf[u][t] = *(const v2f*)(brow + (size_t)t * 16 * K + k + 4 * u);
    }
#pragma unroll
    for (int u = 0; u < U; ++u)
#pragma unroll
      for (int t = 0; t < NT; ++t)
        acc[t] = __builtin_amdgcn_wmma_f32_16x16x4_f32(false, a[u], false,
                                                       bf[u][t], (short)0,
                                                       acc[t], false, false);
  }
  const int rbase = i0 + (lane >> 4) * 8;
#pragma unroll
  for (int t = 0; t < NT; ++t) {
    const int col = j0 + t * 16 + m;
#pragma unroll
    for (int v = 0; v < 8; ++v) C[(size_t)(rbase + v) * ldc + col] = acc[t][v];
  }
}

// ---------- fp32 WMMA GEMM, C[M,N] = A[M,K] * B[K,N]  (row-major) ----------
template <int NT, int U>
__global__ __launch_bounds__(32)
void k_gemm_nn(const float* __restrict__ A, const float* __restrict__ B,
               float* __restrict__ C, int K, int ldb, int ldc) {
  const int j0 = blockIdx.x * (16 * NT);
  const int i0 = blockIdx.y * 16;
  const int lane = threadIdx.x;
  const int m = lane & 15;
  const int koff = (lane >> 4) * 2;
  const float* arow = A + (size_t)(i0 + m) * K + koff;
  v8f acc[NT];
#pragma unroll
  for (int t = 0; t < NT; ++t) acc[t] = (v8f){0.f,0.f,0.f,0.f,0.f,0.f,0.f,0.f};
  for (int k = 0; k < K; k += 4 * U) {
    v2f a[U];
    v2f bf[U][NT];
#pragma unroll
    for (int u = 0; u < U; ++u) {
      a[u] = *(const v2f*)(arow + k + 4 * u);
      const float* b0 = B + (size_t)(k + 4 * u + koff) * ldb + j0 + m;
#pragma unroll
      for (int t = 0; t < NT; ++t) {
        bf[u][t].x = b0[t * 16];
        bf[u][t].y = b0[t * 16 + ldb];
      }
    }
#pragma unroll
    for (int u = 0; u < U; ++u)
#pragma unroll
      for (int t = 0; t < NT; ++t)
        acc[t] = __builtin_amdgcn_wmma_f32_16x16x4_f32(false, a[u], false,
                                                       bf[u][t], (short)0,
                                                       acc[t], false, false);
  }
  const int rbase = i0 + (lane >> 4) * 8;
#pragma unroll
  for (int t = 0; t < NT; ++t) {
    const int col = j0 + t * 16 + m;
#pragma unroll
    for (int v = 0; v < 8; ++v) C[(size_t)(rbase + v) * ldc + col] = acc[t][v];
  }
}

// ---------- stage 4: per-row mobius_matvec tail + bias add + logmap0 scores --
__global__ __launch_bounds__(32)
void k_rowstage(const float* __restrict__ mx, const float* __restrict__ xn,
                const float* __restrict__ eb, const float* __restrict__ awl,
                const float* __restrict__ awr, float* __restrict__ h,
                float* __restrict__ x2o, float* __restrict__ sl,
                float* __restrict__ sr) {
  const int row = blockIdx.x, lane = threadIdx.x;
  float4 mv = ((const float4*)(mx + row * DIM))[lane];
  float mxn = fmaxf(sqrtf(waveSum(dot4(mv, mv))), FEPS);
  float xv = xn[row];
  float r = tanhf(mxn / xv * artanh_c(xv)) / mxn;     // mobius_matvec scale
  float4 hv; hv.x = mv.x * r; hv.y = mv.y * r; hv.z = mv.z * r; hv.w = mv.w * r;
  float n1 = fmaxf(fabsf(r) * mxn, FEPS);             // ||res|| = |tanh(...)|
  float s1 = (n1 > MAXNORM) ? (MAXNORM / n1) : 1.0f;  // proj
  hv.x *= s1; hv.y *= s1; hv.z *= s1; hv.w *= s1;
  // mobius_add(h1, eb)
  float4 e4 = ((const float4*)eb)[lane];
  float x2 = waveSum(dot4(hv, hv));
  float y2 = waveSum(dot4(e4, e4));
  float xy = waveSum(dot4(hv, e4));
  float ca = 1.0f + 2.0f * xy + y2;
  float cb = 1.0f - x2;
  float den = fmaxf(1.0f + 2.0f * xy + x2 * y2, FEPS);
  float4 hm;
  hm.x = (ca * hv.x + cb * e4.x) / den;
  hm.y = (ca * hv.y + cb * e4.y) / den;
  hm.z = (ca * hv.z + cb * e4.z) / den;
  hm.w = (ca * hv.w + cb * e4.w) / den;
  // proj
  float hn = fmaxf(sqrtf(waveSum(dot4(hm, hm))), FEPS);
  float s2 = (hn > MAXNORM) ? (MAXNORM / hn) : 1.0f;
  hm.x *= s2; hm.y *= s2; hm.z *= s2; hm.w *= s2;
  float hna = fmaxf(fminf(hn, MAXNORM), FEPS);
  ((float4*)(h + row * DIM))[lane] = hm;
  // logmap0 scores: h_t = artanh(||h||)/||h|| * h
  float lt = artanh_c(hna) / hna;
  float4 wl = ((const float4*)awl)[lane];
  float4 wr = ((const float4*)awr)[lane];
  float slv = lt * waveSum(dot4(hm, wl));
  float srv = lt * waveSum(dot4(hm, wr));
  if (lane == 0) { x2o[row] = hna * hna; sl[row] = slv; sr[row] = srv; }
}

// ---------- stage 6: per-pair scalars; B overwrites G in place; A row sums ---
__global__ __launch_bounds__(256)
void k_pair(float* G,                              // in: h_i.h_j ; out: B_ij
            const float* __restrict__ x2, const float* __restrict__ sl,
            const float* __restrict__ sr, const float* __restrict__ adj,
            const float* __restrict__ att_b, float* __restrict__ Aout) {
  const int i = blockIdx.x, t = threadIdx.x;
  const float x2i = x2[i], sli = sl[i], ab = att_b[0];
  const float fi = fmaxf(1.0f - x2i, FEPS);        // 2/(sc*lambda_x)
  float accA = 0.0f;
  for (int j = t; j < NROW; j += 256) {
    float dot = G[(size_t)i * NROW + j];
    float y2 = x2[j];
    float den = fmaxf(1.0f - 2.0f * dot + x2i * y2, FEPS);
    float alpha = -(1.0f - 2.0f * dot + y2) / den;
    float beta = (1.0f - x2i) / den;
    float sn2 = alpha * alpha * x2i + 2.0f * alpha * beta * dot + beta * beta * y2;
    float sn = fmaxf(sqrtf(fmaxf(sn2, 0.0f)), FEPS);
    float g = fi * artanh_c(sn) / sn;
    float s = sli + sr[j] + ab;
    float w = adj[(size_t)i * NROW + j] / (1.0f + expf(-s));   // sigmoid*adj
    G[(size_t)i * NROW + j] = w * g * beta;
    accA += w * g * alpha;
  }
  __shared__ float red[8];
  float v = waveSum(accA);
  if ((t & 31) == 0) red[t >> 5] = v;
  __syncthreads();
  if (t == 0) {
    float a = 0.0f;
#pragma unroll
    for (int k = 0; k < 8; ++k) a += red[k];
    Aout[i] = a;
  }
}

// ---------- stage 8: expmap(support, h) -> relu(logmap0) -> proj(expmap0) ----
__global__ __launch_bounds__(32)
void k_final(const float* __restrict__ S, const float* __restrict__ Acoef,
             const float* __restrict__ h, const float* __restrict__ x2,
             float* __restrict__ out) {
  const int row = blockIdx.x, lane = threadIdx.x;
  float4 u = ((const float4*)(S + row * DIM))[lane];
  float4 p = ((const float4*)(h + row * DIM))[lane];
  const float Ai = Acoef[row];
  u.x += Ai * p.x; u.y += Ai * p.y; u.z += Ai * p.z; u.w += Ai * p.w;
  // expmap(u, p)
  float un = fmaxf(sqrtf(waveSum(dot4(u, u))), FEPS);
  float x2i = x2[row];
  float lam = 2.0f / fmaxf(1.0f - x2i, FEPS);
  float ks = tanhf(lam * un * 0.5f) / un;
  float4 sec; sec.x = u.x * ks; sec.y = u.y * ks; sec.z = u.z * ks; sec.w = u.w * ks;
  float xy = waveSum(dot4(p, sec));
  float y2 = waveSum(dot4(sec, sec));
  float ca = 1.0f + 2.0f * xy + y2;
  float cb = 1.0f - x2i;
  float den = fmaxf(1.0f + 2.0f * xy + x2i * y2, FEPS);
  float4 hm;
  hm.x = (ca * p.x + cb * sec.x) / den;
  hm.y = (ca * p.y + cb * sec.y) / den;
  hm.z = (ca * p.z + cb * sec.z) / den;
  hm.w = (ca * p.w + cb * sec.w) / den;
  // proj
  float n = fmaxf(sqrtf(waveSum(dot4(hm, hm))), FEPS);
  float s = (n > MAXNORM) ? (MAXNORM / n) : 1.0f;
  hm.x *= s; hm.y *= s; hm.z *= s; hm.w *= s;
  float na = fmaxf(fminf(n, MAXNORM), FEPS);
  // relu(logmap0)
  float lt = artanh_c(na) / na;
  float4 t4;
  t4.x = fmaxf(hm.x * lt, 0.0f); t4.y = fmaxf(hm.y * lt, 0.0f);
  t4.z = fmaxf(hm.z * lt, 0.0f); t4.w = fmaxf(hm.w * lt, 0.0f);
  // proj(expmap0)
  float nt = fmaxf(sqrtf(waveSum(dot4(t4, t4))), FEPS);
  float eo = tanhf(nt) / nt;
  float4 o; o.x = t4.x * eo; o.y = t4.y * eo; o.z = t4.z * eo; o.w = t4.w * eo;
  float no = fmaxf(sqrtf(waveSum(dot4(o, o))), FEPS);
  float so = (no > MAXNORM) ? (MAXNORM / no) : 1.0f;
  o.x *= so; o.y *= so; o.z *= so; o.w *= so;
  ((float4*)(out + row * DIM))[lane] = o;
}

extern "C" void kernel_launch(void* const* d_in, const int* in_sizes, int n_in,
                              void* d_out, int out_size, void* d_ws, size_t ws_size,
                              hipStream_t stream) {
  const float* x   = (const float*)d_in[0];   // [1024,128]
  const float* adj = (const float*)d_in[1];   // [1024,1024]
  const float* W   = (const float*)d_in[2];   // [128,128]
  const float* b   = (const float*)d_in[3];   // [128]
  const float* awl = (const float*)d_in[4];   // [128]
  const float* awr = (const float*)d_in[5];   // [128]
  const float* ab  = (const float*)d_in[6];   // [1]
  (void)in_sizes; (void)n_in; (void)out_size; (void)ws_size;

  float* ws  = (float*)d_ws;
  float* xp  = ws;                      // N*D   projected x
  float* mxS = xp + NROW * DIM;         // N*D   mx, later reused as support S
  float* h   = mxS + NROW * DIM;        // N*D
  float* G   = h + NROW * DIM;          // N*N   Gram, overwritten in place by B
  float* xn  = G + (size_t)NROW * NROW; // N
  float* x2  = xn + NROW;               // N
  float* sl  = x2 + NROW;               // N
  float* sr  = sl + NROW;               // N
  float* Ac  = sr + NROW;               // N
  float* eb  = Ac + NROW;               // D

  k_proj_x<<<NROW, 32, 0, stream>>>(x, xp, xn);
  k_bias<<<1, 32, 0, stream>>>(b, eb);
  // mx = xp @ W^T   (M=1024, N=128, K=128): 2 x 64 waves, 4 tiles each
  k_gemm_nt<4, 4><<<dim3(DIM / 64, NROW / 16), 32, 0, stream>>>(xp, W, mxS,
                                                                DIM, DIM);
  k_rowstage<<<NROW, 32, 0, stream>>>(mxS, xn, eb, awl, awr, h, x2, sl, sr);
  // G = h @ h^T     (M=N=1024, K=128): 16 x 64 waves, 4 tiles each
  k_gemm_nt<4, 4><<<dim3(NROW / 64, NROW / 16), 32, 0, stream>>>(h, h, G,
                                                                 DIM, NROW);
  k_pair<<<NROW, 256, 0, stream>>>(G, x2, sl, sr, adj, ab, Ac);
  // S = B @ h       (M=1024, N=128, K=1024): 2 x 64 waves, 4 tiles each
  k_gemm_nn<4, 2><<<dim3(DIM / 64, NROW / 16), 32, 0, stream>>>(G, h, mxS,
                                                                NROW, DIM, DIM);
  k_final<<<NROW, 32, 0, stream>>>(mxS, Ac, h, x2, (float*)d_out);
}